// Render_9457517986250
// MI455X (gfx1250) — compile-verified
//
#include <hip/hip_runtime.h>

typedef float v2f __attribute__((ext_vector_type(2)));
typedef float v8f __attribute__((ext_vector_type(8)));
typedef int   v4i __attribute__((vector_size(16)));   // matches builtin diag type

#define NTRI   1024
#define NPAIR  512
#define SIZE   512
#define TEXN   1024
#define TEXSZ  (1024 * 1024)
#define CHUNK  64                 // triangle pairs staged per LDS chunk
#define NCHUNK (NPAIR / CHUNK)

#if defined(__has_builtin)
#if __has_builtin(__builtin_amdgcn_global_load_async_to_lds_b128) && \
    __has_builtin(__builtin_amdgcn_global_load_async_to_lds_b32)  && \
    __has_builtin(__builtin_amdgcn_s_wait_asynccnt)
#define HAVE_ASYNC_LDS 1
#endif
#endif

// exact pointer types for the async-to-LDS builtins (from hipcc diagnostics:
// param0 = T addrspace(1)*  [printed as "__device__"], param1 = T addrspace(3)*)
typedef __attribute__((address_space(1))) v4i* g_v4i_p;
typedef __attribute__((address_space(3))) v4i* l_v4i_p;
typedef __attribute__((address_space(1))) int* g_i32_p;
typedef __attribute__((address_space(3))) int* l_i32_p;

__device__ __forceinline__ int nidx_clip(float v) {
    v = fminf(fmaxf(v, -1.0f), 1.0f);
    return (int)floorf((v + 1.0f) * 0.5f * (float)SIZE);
}

// ---------------------------------------------------------------------------
// Kernel 1: per-triangle affine plane setup.
// coef layout: [pair p][half][row r (0..15)][2 floats]  (256 B per pair)
//   half=0 -> (ax, ay) = A-matrix K=0,1 (lanes 0-15)
//   half=1 -> (c , 0 ) = A-matrix K=2,3 (lanes 16-31)
//   rows 0-5  : e0,e1,e2,z,u,v planes of triangle 2p
//   rows 6,7  : y-box planes  f6 = j - ymin + 0.5 ,  f7 = ymax - 0.5 - j
//   rows 8-15 : same for triangle 2p+1
// box[t] = xmin | (xmax << 16)           (y handled by rows 6,7)
// ---------------------------------------------------------------------------
__global__ void setup_coefs_kernel(const float* __restrict__ tris,
                                   const float* __restrict__ uvs,
                                   float* __restrict__ coef,
                                   int* __restrict__ box)
{
    int t = blockIdx.x * blockDim.x + threadIdx.x;
    if (t >= NTRI) return;

    const float* tv = tris + t * 9;
    float v0x = tv[0], v0y = tv[1], z0 = tv[2];
    float v1x = tv[3], v1y = tv[4], z1 = tv[5];
    float v2x = tv[6], v2y = tv[7], z2 = tv[8];
    const float* up = uvs + t * 6;
    float u0 = up[0] * 2.f - 1.f, q0 = up[1] * 2.f - 1.f;
    float u1 = up[2] * 2.f - 1.f, q1 = up[3] * 2.f - 1.f;
    float u2 = up[4] * 2.f - 1.f, q2 = up[5] * 2.f - 1.f;

    // Edge planes: pAB, pCB, pCA (each = ax*x + ay*y + c)
    float ax0 = v0y - v1y, ay0 = v1x - v0x;
    float c0  = -(v1x * ax0 + v1y * ay0);
    float ax1 = v1y - v2y, ay1 = v2x - v1x;
    float c1  = -(v2x * ax1 + v2y * ay1);
    float ax2 = v2y - v0y, ay2 = v0x - v2x;
    float c2  = -(v0x * ax2 + v0y * ay2);

    float w = (v1x - v0x) * (v2y - v0y) - (v1y - v0y) * (v2x - v0x);
    bool valid = (w >= 1e-9f);
    float inv = 1.0f / (valid ? w : 1.0f);
    if (!valid) { ax0 = 0.f; ay0 = 0.f; c0 = -1.f; }  // kill coverage everywhere

    // Interpolation planes: f(p) = f2 + (pCB*(f0-f2) + pCA*(f1-f2)) / w
    float dz0 = z0 - z2, dz1 = z1 - z2;
    float zax = (ax1 * dz0 + ax2 * dz1) * inv;
    float zay = (ay1 * dz0 + ay2 * dz1) * inv;
    float zc  = z2 + (c1 * dz0 + c2 * dz1) * inv;

    float du0 = u0 - u2, du1 = u1 - u2;
    float uax = (ax1 * du0 + ax2 * du1) * inv;
    float uay = (ay1 * du0 + ay2 * du1) * inv;
    float uc  = u2 + (c1 * du0 + c2 * du1) * inv;

    float dq0 = q0 - q2, dq1 = q1 - q2;
    float qax = (ax1 * dq0 + ax2 * dq1) * inv;
    float qay = (ay1 * dq0 + ay2 * dq1) * inv;
    float qc  = q2 + (c1 * dq0 + c2 * dq1) * inv;

    float minx = fminf(v0x, fminf(v1x, v2x));
    float miny = fminf(v0y, fminf(v1y, v2y));
    float maxx = fmaxf(v0x, fmaxf(v1x, v2x));
    float maxy = fmaxf(v0y, fmaxf(v1y, v2y));
    int xmin = nidx_clip(minx), ymin = nidx_clip(miny);
    int xmax = nidx_clip(maxx), ymax = nidx_clip(maxy);

    // y-box planes: j = 255.5*py + 255.5 exactly (integer margins 0.5)
    float rax[8] = { ax0, ax1, ax2, zax, uax, qax, 0.0f,    0.0f };
    float ray[8] = { ay0, ay1, ay2, zay, uay, qay, 255.5f, -255.5f };
    float rc [8] = { c0,  c1,  c2,  zc,  uc,  qc,
                     256.0f - (float)ymin, (float)ymax - 256.0f };

    int   p    = t >> 1;
    int   hrow = (t & 1) * 8;
    float* base = coef + p * 64;
#pragma unroll
    for (int f = 0; f < 8; ++f) {
        int r = hrow + f;
        base[r * 2 + 0]      = rax[f];
        base[r * 2 + 1]      = ray[f];
        base[32 + r * 2 + 0] = rc[f];
        base[32 + r * 2 + 1] = 0.f;
    }
    box[t] = xmin | (xmax << 16);
}

// ---------------------------------------------------------------------------
// Kernel 2: WMMA rasterization. Each wave owns 16 pixels (row i, cols j..j+15).
// One V_WMMA_F32_16X16X4_F32 evaluates 8 planes of 2 triangles over those
// pixels. Triangle-pair stream is staged per-block into LDS with gfx1250
// async global->LDS copies (8 waves share each chunk).
// ---------------------------------------------------------------------------
__global__ __launch_bounds__(256) void render_kernel(
        const float* __restrict__ coef,
        const int*   __restrict__ box,
        const float* __restrict__ uvmap,
        float*       __restrict__ out)
{
    __shared__ float scoef[CHUNK * 64];   // 16 KB
    __shared__ int   sbox[CHUNK * 2];     // 512 B

    const int tid  = threadIdx.x;
    const int lane = tid & 31;
    const int wave = tid >> 5;
    const int sub  = lane & 15;
    const int hi   = lane >> 4;           // 0: triangle 2p, 1: triangle 2p+1
    const int i    = blockIdx.x * 8 + wave;
    const int j    = blockIdx.y * 16 + sub;

    const float px = fmaf((float)i, 2.0f / 511.0f, -1.0f);
    const float py = fmaf((float)j, 2.0f / 511.0f, -1.0f);

    // B matrix (4x16), K rows = [x, y, 1, 0]
    v2f bmat;
    bmat.x = hi ? 1.0f : px;
    bmat.y = hi ? 0.0f : py;

    float best_z = -3.402823466e+38f;
    float best_u = 0.0f, best_q = 0.0f;
    int   best_idx = -1;

    for (int c = 0; c < NCHUNK; ++c) {
        __syncthreads();                          // previous chunk consumed
        const float* gsrc = coef + (size_t)c * CHUNK * 64;
#if HAVE_ASYNC_LDS
#pragma unroll
        for (int k = 0; k < 4; ++k) {             // 256 thr * 16 floats = chunk
            __builtin_amdgcn_global_load_async_to_lds_b128(
                (g_v4i_p)(gsrc + tid * 16 + k * 4),
                (l_v4i_p)(scoef + tid * 16 + k * 4), 0, 0);
        }
        if (tid < CHUNK * 2) {
            __builtin_amdgcn_global_load_async_to_lds_b32(
                (g_i32_p)(box + c * CHUNK * 2 + tid),
                (l_i32_p)(sbox + tid), 0, 0);
        }
        __builtin_amdgcn_s_wait_asynccnt(0);
#else
#pragma unroll
        for (int k = 0; k < 4; ++k)
            ((float4*)scoef)[tid * 4 + k] = ((const float4*)gsrc)[tid * 4 + k];
        if (tid < CHUNK * 2) sbox[tid] = box[c * CHUNK * 2 + tid];
#endif
        if (c + 1 < NCHUNK)
            __builtin_prefetch(coef + (size_t)(c + 1) * CHUNK * 64 + tid * 16, 0, 1);
        __syncthreads();

#pragma unroll 4
        for (int q = 0; q < CHUNK; ++q) {
            v2f amat = *(const v2f*)(scoef + q * 64 + hi * 32 + sub * 2);  // ds b64
            int bb   = sbox[q * 2 + hi];                                   // ds b32

            v8f czero = {};
            v8f d = __builtin_amdgcn_wmma_f32_16x16x4_f32(
                false, amat, false, bmat, (short)0, czero, false, false);

            // d[0..2]=edges, d[3]=z, d[4]=u, d[5]=v, d[6..7]=y-box planes
            float m = fminf(fminf(d[0], d[1]),
                            fminf(d[2], fminf(d[6], d[7])));   // min3-able
            int xmin = bb & 0xffff;
            int xmax = (int)(((unsigned)bb) >> 16);
            bool take = (m > 0.0f) & (i >= xmin) & (i < xmax) & (d[3] >= best_z);
            int tri = (c * CHUNK + q) * 2 + hi;
            best_z   = take ? d[3] : best_z;
            best_u   = take ? d[4] : best_u;
            best_q   = take ? d[5] : best_q;
            best_idx = take ? tri : best_idx;
        }
    }

    // Merge the two half-wave triangle streams (lexicographic (z, idx) max).
    float oz = __shfl_xor(best_z, 16, 32);
    float ou = __shfl_xor(best_u, 16, 32);
    float oq = __shfl_xor(best_q, 16, 32);
    int   oi = __shfl_xor(best_idx, 16, 32);
    bool adopt = (oz > best_z) || ((oz == best_z) && (oi > best_idx));
    best_z   = adopt ? oz : best_z;
    best_u   = adopt ? ou : best_u;
    best_q   = adopt ? oq : best_q;
    best_idx = adopt ? oi : best_idx;

    if (hi == 0) {
        float r = 0.f, g = 0.f, b = 0.f, a = 0.f;
        if (best_idx >= 0) {
            // bilinear grid_sample, zero padding, align_corners=False
            float x = ((best_u + 1.0f) * (float)TEXN - 1.0f) * 0.5f;
            float y = ((best_q + 1.0f) * (float)TEXN - 1.0f) * 0.5f;
            float x0f = floorf(x), y0f = floorf(y);
            float wx1 = x - x0f, wx0 = 1.0f - wx1;
            float wy1 = y - y0f, wy0 = 1.0f - wy1;
            int x0 = (int)x0f, y0 = (int)y0f;
            int x1 = x0 + 1,   y1 = y0 + 1;
#pragma unroll
            for (int cy = 0; cy < 2; ++cy) {
#pragma unroll
                for (int cx = 0; cx < 2; ++cx) {
                    int xi = cx ? x1 : x0;
                    int yi = cy ? y1 : y0;
                    float wgt = (cx ? wx1 : wx0) * (cy ? wy1 : wy0);
                    bool v = (xi >= 0) & (xi <= TEXN - 1) & (yi >= 0) & (yi <= TEXN - 1);
                    int xc = xi < 0 ? 0 : (xi > TEXN - 1 ? TEXN - 1 : xi);
                    int yc = yi < 0 ? 0 : (yi > TEXN - 1 ? TEXN - 1 : yi);
                    float wv = v ? wgt : 0.0f;
                    size_t o = (size_t)yc * TEXN + xc;
                    r = fmaf(uvmap[o], wv, r);
                    g = fmaf(uvmap[TEXSZ + o], wv, g);
                    b = fmaf(uvmap[2 * TEXSZ + o], wv, b);
                }
            }
            a = 1.0f;
        }
        float4 rgba = make_float4(r, g, b, a);
        *(float4*)(out + ((size_t)i * SIZE + j) * 4) = rgba;
    }
}

// ---------------------------------------------------------------------------
extern "C" void kernel_launch(void* const* d_in, const int* in_sizes, int n_in,
                              void* d_out, int out_size, void* d_ws, size_t ws_size,
                              hipStream_t stream) {
    const float* tris  = (const float*)d_in[0];   // (1024,3,3) f32
    const float* uvs   = (const float*)d_in[1];   // (1024,3,2) f32
    const float* uvmap = (const float*)d_in[2];   // (3,1024,1024) f32
    // d_in[3] = size (=512), compile-time constant here

    float* coef = (float*)d_ws;                                   // 128 KB
    int*   box  = (int*)((char*)d_ws + (size_t)NPAIR * 64 * 4);   // 4 KB

    setup_coefs_kernel<<<dim3(4), dim3(256), 0, stream>>>(tris, uvs, coef, box);

    dim3 grid(SIZE / 8, SIZE / 16);   // 64 x 32 blocks, 8 waves/block
    render_kernel<<<grid, dim3(256), 0, stream>>>(coef, box, uvmap, (float*)d_out);
}